// SR_GNN_attn_86998857548160
// MI455X (gfx1250) — compile-verified
//
#include <hip/hip_runtime.h>
#include <hip/hip_bf16.h>

#define N_NODES 49152
#define N_EDGES 98304
#define N_BATCH 4096
#define HDIM    256
#define EDIM    64
#define NITEMS  50000
#define KX      321     // 1 + 5*64
#define KXP     324     // padded to multiple of 4
#define NTILES  3125    // 50000 / 16

typedef float v2f __attribute__((ext_vector_type(2)));
typedef float v8f __attribute__((ext_vector_type(8)));

static __device__ __forceinline__ v8f wmma4(v2f a, v2f b, v8f c) {
  // V_WMMA_F32_16X16X4_F32 : D = A(16x4) * B(4x16) + C, f32 accumulate
  return __builtin_amdgcn_wmma_f32_16x16x4_f32(false, a, false, b, (short)0, c,
                                               false, false);
}

// ---------------------------------------------------------------------------
// h = [price | emb_cat | emb_sub | emb_el | emb_brand | emb_item] @ Wmsg + b
// one block = 16 rows; 8 waves x 2 column tiles of 16 -> all 256 cols
// ---------------------------------------------------------------------------
__global__ __launch_bounds__(256) void k_embed_gemm(
    const int* __restrict__ cat, const int* __restrict__ sub,
    const int* __restrict__ el,  const int* __restrict__ br,
    const int* __restrict__ pid, const float* __restrict__ price,
    const float* __restrict__ Ecat, const float* __restrict__ Esub,
    const float* __restrict__ Eel,  const float* __restrict__ Ebr,
    const float* __restrict__ Eitem,
    const float* __restrict__ Wmsg, const float* __restrict__ bmsg,
    float* __restrict__ h)
{
  __shared__ float As[16 * KXP];
  const int tid = threadIdx.x;
  const int rowBase = blockIdx.x * 16;
  for (int idx = tid; idx < 16 * KXP; idx += 256) {
    int r = idx / KXP, k = idx % KXP;
    int row = rowBase + r;
    float v = 0.0f;
    if (k == 0)              v = price[row];
    else if (k < 1 + EDIM)   v = Ecat [cat[row] * EDIM + (k - 1)];
    else if (k < 1 + 2*EDIM) v = Esub [sub[row] * EDIM + (k - 65)];
    else if (k < 1 + 3*EDIM) v = Eel  [el [row] * EDIM + (k - 129)];
    else if (k < 1 + 4*EDIM) v = Ebr  [br [row] * EDIM + (k - 193)];
    else if (k < KX)         v = Eitem[pid[row] * EDIM + (k - 257)];
    As[idx] = v;
  }
  __syncthreads();
  const int wave = tid >> 5, lane = tid & 31;
  const int hl = lane & 15;
  const int kb = (lane < 16) ? 0 : 2;
  const int mb = (lane < 16) ? 0 : 8;
  const int n0 = wave * 16 + hl;
  const int n1 = (wave + 8) * 16 + hl;
  v8f c0 = {}; v8f c1 = {};
  for (int ks = 0; ks < KXP / 4; ++ks) {
    int k = 4 * ks + kb;
    v2f a; a.x = As[hl * KXP + k]; a.y = As[hl * KXP + k + 1];
    v2f b0, b1;
    b0.x = (k     < KX) ? Wmsg[(k    ) * HDIM + n0] : 0.0f;
    b0.y = (k + 1 < KX) ? Wmsg[(k + 1) * HDIM + n0] : 0.0f;
    b1.x = (k     < KX) ? Wmsg[(k    ) * HDIM + n1] : 0.0f;
    b1.y = (k + 1 < KX) ? Wmsg[(k + 1) * HDIM + n1] : 0.0f;
    c0 = wmma4(a, b0, c0);
    c1 = wmma4(a, b1, c1);
  }
  for (int j = 0; j < 8; ++j) {
    int row = rowBase + mb + j;
    h[row * HDIM + n0] = c0[j] + bmsg[n0];
    h[row * HDIM + n1] = c1[j] + bmsg[n1];
  }
}

// ---------------------------------------------------------------------------
// edge scatter: msum[dst] += h[src]; cnt[dst] += 1   (one block per edge)
// ---------------------------------------------------------------------------
__global__ __launch_bounds__(256) void k_edge_scatter(
    const int* __restrict__ ei, const float* __restrict__ h,
    float* msum, float* cnt)
{
  const int e = blockIdx.x, t = threadIdx.x;
  const int src = ei[e], dst = ei[N_EDGES + e];
  atomicAdd(&msum[dst * HDIM + t], h[src * HDIM + t]);
  if (t == 0) atomicAdd(&cnt[dst], 1.0f);
}

__global__ __launch_bounds__(256) void k_normalize(float* m, const float* __restrict__ cnt)
{
  const int idx = blockIdx.x * 256 + threadIdx.x;   // grid = N blocks
  const int i = idx >> 8;
  float c = cnt[i]; c = (c > 1.0f) ? c : 1.0f;
  m[idx] = m[idx] / c;
}

// ---------------------------------------------------------------------------
// fused GRU cell: gi = m@Wih+bih, gh = h@Whh+bhh, gate, h_new -> hnew (== m)
// block = 16 rows; wave holds 6 WMMA accumulators (r,z,n for both GEMMs)
// ---------------------------------------------------------------------------
__global__ __launch_bounds__(256) void k_gru(
    const float* m, const float* __restrict__ h,
    const float* __restrict__ Wih, const float* __restrict__ bih,
    const float* __restrict__ Whh, const float* __restrict__ bhh,
    float* hnew)
{
  __shared__ float Ms[16 * 260];
  __shared__ float Hs[16 * 260];
  const int tid = threadIdx.x;
  const int rowBase = blockIdx.x * 16;
  for (int idx = tid; idx < 16 * HDIM; idx += 256) {
    int r = idx >> 8, c = idx & 255;
    Ms[r * 260 + c] = m[(rowBase + r) * HDIM + c];
    Hs[r * 260 + c] = h[(rowBase + r) * HDIM + c];
  }
  __syncthreads();
  const int wave = tid >> 5, lane = tid & 31;
  const int hl = lane & 15;
  const int kb = (lane < 16) ? 0 : 2;
  const int mb = (lane < 16) ? 0 : 8;
  for (int half = 0; half < 2; ++half) {
    const int col = (wave + 8 * half) * 16 + hl;    // 0..255
    v8f ir = {}, iz = {}, in_ = {}, hr = {}, hz = {}, hn = {};
    for (int ks = 0; ks < HDIM / 4; ++ks) {
      int k = 4 * ks + kb;
      v2f am; am.x = Ms[hl * 260 + k]; am.y = Ms[hl * 260 + k + 1];
      v2f ah; ah.x = Hs[hl * 260 + k]; ah.y = Hs[hl * 260 + k + 1];
      v2f b;
      b.x = Wih[k * 768 + col];        b.y = Wih[(k + 1) * 768 + col];
      ir  = wmma4(am, b, ir);
      b.x = Wih[k * 768 + 256 + col];  b.y = Wih[(k + 1) * 768 + 256 + col];
      iz  = wmma4(am, b, iz);
      b.x = Wih[k * 768 + 512 + col];  b.y = Wih[(k + 1) * 768 + 512 + col];
      in_ = wmma4(am, b, in_);
      b.x = Whh[k * 768 + col];        b.y = Whh[(k + 1) * 768 + col];
      hr  = wmma4(ah, b, hr);
      b.x = Whh[k * 768 + 256 + col];  b.y = Whh[(k + 1) * 768 + 256 + col];
      hz  = wmma4(ah, b, hz);
      b.x = Whh[k * 768 + 512 + col];  b.y = Whh[(k + 1) * 768 + 512 + col];
      hn  = wmma4(ah, b, hn);
    }
    const float bir = bih[col], biz = bih[256 + col], bin = bih[512 + col];
    const float bhr = bhh[col], bhz = bhh[256 + col], bhn = bhh[512 + col];
    for (int j = 0; j < 8; ++j) {
      int mrow = mb + j;
      float i_r = ir[j] + bir, i_z = iz[j] + biz, i_n = in_[j] + bin;
      float h_r = hr[j] + bhr, h_z = hz[j] + bhz, h_n = hn[j] + bhn;
      float r = 1.0f / (1.0f + __expf(-(i_r + h_r)));
      float z = 1.0f / (1.0f + __expf(-(i_z + h_z)));
      float n = tanhf(i_n + r * h_n);
      float ho = Hs[mrow * 260 + col];
      hnew[(rowBase + mrow) * HDIM + col] = (1.0f - z) * n + z * ho;
    }
  }
}

// ---------------------------------------------------------------------------
// hout = hin @ W(256x256) + bias   (attention linear)
// ---------------------------------------------------------------------------
__global__ __launch_bounds__(256) void k_attn(
    const float* __restrict__ hin, const float* __restrict__ W,
    const float* __restrict__ bias, float* __restrict__ hout)
{
  __shared__ float As[16 * 260];
  const int tid = threadIdx.x;
  const int rowBase = blockIdx.x * 16;
  for (int idx = tid; idx < 16 * HDIM; idx += 256) {
    int r = idx >> 8, c = idx & 255;
    As[r * 260 + c] = hin[(rowBase + r) * HDIM + c];
  }
  __syncthreads();
  const int wave = tid >> 5, lane = tid & 31;
  const int hl = lane & 15;
  const int kb = (lane < 16) ? 0 : 2;
  const int mb = (lane < 16) ? 0 : 8;
  const int n0 = wave * 16 + hl, n1 = (wave + 8) * 16 + hl;
  v8f c0 = {}, c1 = {};
  for (int ks = 0; ks < HDIM / 4; ++ks) {
    int k = 4 * ks + kb;
    v2f a; a.x = As[hl * 260 + k]; a.y = As[hl * 260 + k + 1];
    v2f b0, b1;
    b0.x = W[k * HDIM + n0]; b0.y = W[(k + 1) * HDIM + n0];
    b1.x = W[k * HDIM + n1]; b1.y = W[(k + 1) * HDIM + n1];
    c0 = wmma4(a, b0, c0);
    c1 = wmma4(a, b1, c1);
  }
  for (int j = 0; j < 8; ++j) {
    int row = rowBase + mb + j;
    hout[row * HDIM + n0] = c0[j] + bias[n0];
    hout[row * HDIM + n1] = c1[j] + bias[n1];
  }
}

// ---------------------------------------------------------------------------
// segment helpers
// ---------------------------------------------------------------------------
__global__ void k_fill_f32(float* p, int n, float v) {
  int i = blockIdx.x * 256 + threadIdx.x; if (i < n) p[i] = v;
}
__global__ void k_fill_i32(int* p, int n, int v) {
  int i = blockIdx.x * 256 + threadIdx.x; if (i < n) p[i] = v;
}
__global__ void k_lastidx(const int* __restrict__ batch, int* lastI) {
  int i = blockIdx.x * 256 + threadIdx.x;
  if (i < N_NODES) atomicMax(&lastI[batch[i]], i);
}

// scores[i] = sigmoid(h_lt[i] + h_lt[last_idx[batch[i]]]) . Wsc + bsc
__global__ __launch_bounds__(256) void k_score(
    const float* __restrict__ hlt, const int* __restrict__ batch,
    const int* __restrict__ lastI, const float* __restrict__ Wsc,
    const float* __restrict__ bsc, float* __restrict__ s)
{
  __shared__ float red[256];
  const int i = blockIdx.x, t = threadIdx.x;
  const int li = lastI[batch[i]];
  float v = hlt[i * HDIM + t] + hlt[li * HDIM + t];
  v = 1.0f / (1.0f + __expf(-v));
  red[t] = v * Wsc[t];
  __syncthreads();
  for (int st = 128; st > 0; st >>= 1) {
    if (t < st) red[t] += red[t + st];
    __syncthreads();
  }
  if (t == 0) s[i] = red[0] + bsc[0];
}

__global__ void k_smax(const float* __restrict__ s, const int* __restrict__ batch,
                       float* smax) {
  int i = blockIdx.x * 256 + threadIdx.x;
  if (i < N_NODES) atomicMax(&smax[batch[i]], s[i]);
}
__global__ void k_expdenom(float* s, const int* __restrict__ batch,
                           const float* __restrict__ smax, float* denom) {
  int i = blockIdx.x * 256 + threadIdx.x;
  if (i < N_NODES) {
    float e = __expf(s[i] - smax[batch[i]]);
    s[i] = e;
    atomicAdd(&denom[batch[i]], e);
  }
}
__global__ __launch_bounds__(256) void k_sess(
    const float* __restrict__ hlt, const float* __restrict__ s,
    const int* __restrict__ batch, const float* __restrict__ denom,
    float* sess)
{
  const int i = blockIdx.x, t = threadIdx.x;
  const int b = batch[i];
  const float w = s[i] / denom[b];
  atomicAdd(&sess[b * HDIM + t], w * hlt[i * HDIM + t]);
}

// ---------------------------------------------------------------------------
// out = sess(4096x256) @ Wfc(256x50000) + bfc   -- the dominant GEMM
// grid: x = col blocks (16 tiles each), y = row tiles. Wfc (51MB) lives in L2.
// ---------------------------------------------------------------------------
__global__ __launch_bounds__(256) void k_fc(
    const float* __restrict__ sess, const float* __restrict__ Wfc,
    const float* __restrict__ bfc, float* __restrict__ out)
{
  __shared__ float As[16 * 260];
  const int tid = threadIdx.x;
  const int rowBase = blockIdx.y * 16;
  const int ctBase  = blockIdx.x * 16;
  for (int idx = tid; idx < 16 * HDIM; idx += 256) {
    int r = idx >> 8, c = idx & 255;
    As[r * 260 + c] = sess[(rowBase + r) * HDIM + c];
  }
  __syncthreads();
  const int wave = tid >> 5, lane = tid & 31;
  const int hl = lane & 15;
  const int kb = (lane < 16) ? 0 : 2;
  const int mb = (lane < 16) ? 0 : 8;
  const int ct0 = ctBase + wave, ct1 = ctBase + wave + 8;
  const int n0 = ct0 * 16 + hl, n1 = ct1 * 16 + hl;
  const bool v0 = ct0 < NTILES, v1 = ct1 < NTILES;  // wave-uniform
  v8f c0 = {}, c1 = {};
  if (v0 && v1) {
    for (int ks = 0; ks < HDIM / 4; ++ks) {
      int k = 4 * ks + kb;
      v2f a; a.x = As[hl * 260 + k]; a.y = As[hl * 260 + k + 1];
      v2f b0, b1;
      b0.x = Wfc[k * NITEMS + n0]; b0.y = Wfc[(k + 1) * NITEMS + n0];
      b1.x = Wfc[k * NITEMS + n1]; b1.y = Wfc[(k + 1) * NITEMS + n1];
      if (ks + 2 < HDIM / 4)
        __builtin_prefetch(&Wfc[(k + 8) * NITEMS + n0], 0, 1);
      c0 = wmma4(a, b0, c0);
      c1 = wmma4(a, b1, c1);
    }
    for (int j = 0; j < 8; ++j) {
      long row = rowBase + mb + j;
      out[row * NITEMS + n0] = c0[j] + bfc[n0];
      out[row * NITEMS + n1] = c1[j] + bfc[n1];
    }
  } else if (v0) {
    for (int ks = 0; ks < HDIM / 4; ++ks) {
      int k = 4 * ks + kb;
      v2f a; a.x = As[hl * 260 + k]; a.y = As[hl * 260 + k + 1];
      v2f b0;
      b0.x = Wfc[k * NITEMS + n0]; b0.y = Wfc[(k + 1) * NITEMS + n0];
      c0 = wmma4(a, b0, c0);
    }
    for (int j = 0; j < 8; ++j) {
      long row = rowBase + mb + j;
      out[row * NITEMS + n0] = c0[j] + bfc[n0];
    }
  }
}

// ---------------------------------------------------------------------------
extern "C" void kernel_launch(void* const* d_in, const int* in_sizes, int n_in,
                              void* d_out, int out_size, void* d_ws, size_t ws_size,
                              hipStream_t stream)
{
  const int*   cat   = (const int*)d_in[0];
  const int*   sub   = (const int*)d_in[1];
  const int*   el    = (const int*)d_in[2];
  const int*   br    = (const int*)d_in[3];
  const int*   pid   = (const int*)d_in[4];
  const float* price = (const float*)d_in[5];
  const int*   ei    = (const int*)d_in[6];
  const int*   batch = (const int*)d_in[7];
  const float* Ecat  = (const float*)d_in[8];
  const float* Esub  = (const float*)d_in[9];
  const float* Eel   = (const float*)d_in[10];
  const float* Ebr   = (const float*)d_in[11];
  const float* Eitem = (const float*)d_in[12];
  const float* Wmsg  = (const float*)d_in[13];
  const float* bmsg  = (const float*)d_in[14];
  const float* Wih   = (const float*)d_in[15];
  const float* bih   = (const float*)d_in[16];
  const float* Whh   = (const float*)d_in[17];
  const float* bhh   = (const float*)d_in[18];
  const float* Wattn = (const float*)d_in[19];
  const float* battn = (const float*)d_in[20];
  const float* Wsc   = (const float*)d_in[21];
  const float* bsc   = (const float*)d_in[22];
  const float* Wfc   = (const float*)d_in[23];
  const float* bfc   = (const float*)d_in[24];
  float* out = (float*)d_out;

  // workspace layout (~105 MB)
  float* ws    = (float*)d_ws;
  float* h     = ws;                              // N*256 (later h_lt)
  float* m     = h + (size_t)N_NODES * HDIM;      // N*256 (later h_new)
  float* cnt   = m + (size_t)N_NODES * HDIM;      // N
  float* s     = cnt + N_NODES;                   // N
  float* smax  = s + N_NODES;                     // B
  float* denom = smax + N_BATCH;                  // B
  float* sess  = denom + N_BATCH;                 // B*256
  int*   lastI = (int*)(sess + (size_t)N_BATCH * HDIM);   // B

  dim3 b256(256);
  // 1) h = gathered-x @ Wmsg + bmsg
  k_embed_gemm<<<dim3(N_NODES / 16), b256, 0, stream>>>(
      cat, sub, el, br, pid, price, Ecat, Esub, Eel, Ebr, Eitem, Wmsg, bmsg, h);
  // 2) mean aggregation over edges
  k_fill_f32<<<dim3((N_NODES * HDIM + 255) / 256), b256, 0, stream>>>(m, N_NODES * HDIM, 0.0f);
  k_fill_f32<<<dim3((N_NODES + 255) / 256), b256, 0, stream>>>(cnt, N_NODES, 0.0f);
  k_edge_scatter<<<dim3(N_EDGES), b256, 0, stream>>>(ei, h, m, cnt);
  k_normalize<<<dim3(N_NODES), b256, 0, stream>>>(m, cnt);
  // 3) GRU cell (h_new overwrites m)
  k_gru<<<dim3(N_NODES / 16), b256, 0, stream>>>(m, h, Wih, bih, Whh, bhh, m);
  // 4) h_lt = h_new @ Wattn + battn (overwrites h)
  k_attn<<<dim3(N_NODES / 16), b256, 0, stream>>>(m, Wattn, battn, h);
  // 5) per-session last index
  k_fill_i32<<<dim3((N_BATCH + 255) / 256), b256, 0, stream>>>(lastI, N_BATCH, 0);
  k_lastidx<<<dim3((N_NODES + 255) / 256), b256, 0, stream>>>(batch, lastI);
  // 6) attention scores + segment softmax
  k_score<<<dim3(N_NODES), b256, 0, stream>>>(h, batch, lastI, Wsc, bsc, s);
  k_fill_f32<<<dim3((N_BATCH + 255) / 256), b256, 0, stream>>>(smax, N_BATCH, -3.0e38f);
  k_smax<<<dim3((N_NODES + 255) / 256), b256, 0, stream>>>(s, batch, smax);
  k_fill_f32<<<dim3((N_BATCH + 255) / 256), b256, 0, stream>>>(denom, N_BATCH, 0.0f);
  k_expdenom<<<dim3((N_NODES + 255) / 256), b256, 0, stream>>>(s, batch, smax, denom);
  // 7) weighted session embedding
  k_fill_f32<<<dim3((N_BATCH * HDIM + 255) / 256), b256, 0, stream>>>(sess, N_BATCH * HDIM, 0.0f);
  k_sess<<<dim3(N_NODES), b256, 0, stream>>>(h, s, batch, denom, sess);
  // 8) out = sess @ Wfc + bfc
  k_fc<<<dim3((NTILES + 15) / 16, N_BATCH / 16), b256, 0, stream>>>(sess, Wfc, bfc, out);
}